// DEnKF_89464168776337
// MI455X (gfx1250) — compile-verified
//
#include <hip/hip_runtime.h>
#include <hip/hip_bf16.h>
#include <math.h>

// ---------------------------------------------------------------------------
// DEnKF forward for MI455X (gfx1250, wave32, WMMA).
// Dense/flipout layers: block-tiled WMMA GEMM (V_WMMA_F32_16X16X4_F32),
// weights staged to LDS (async-to-LDS when available), 2-8 WMMA per A load.
// ---------------------------------------------------------------------------

typedef __attribute__((ext_vector_type(2))) float v2f;
typedef __attribute__((ext_vector_type(8))) float v8f;
typedef int v4i_vs __attribute__((vector_size(16)));  // matches async builtin param

#define ENS 128

// ---------------- deterministic hash RNG (stand-in for jax threefry) -------
__device__ __forceinline__ uint32_t hash32(uint32_t x) {
  x ^= x >> 16; x *= 0x7feb352du;
  x ^= x >> 15; x *= 0x846ca68bu;
  x ^= x >> 16;
  return x;
}
__device__ __forceinline__ float rsign(uint32_t seed, uint32_t idx) {
  uint32_t h = hash32(seed ^ (idx * 0x9e3779b9u));
  union { uint32_t u; float f; } v;
  v.u = 0x3f800000u | (h << 31);
  return v.f;
}
__device__ __forceinline__ float nrm(uint32_t seed, uint32_t idx) {
  uint32_t h1 = hash32(seed ^ ((idx * 2u + 1u) * 0x9e3779b9u));
  uint32_t h2 = hash32(seed ^ ((idx * 2u + 2u) * 0x85ebca6bu));
  float u1 = (float)(h1 | 1u) * 2.3283064365386963e-10f;
  float u2 = (float)h2 * 2.3283064365386963e-10f;
  return sqrtf(-2.0f * __logf(u1)) * __cosf(6.28318530717958647f * u2);
}

__global__ void noise_scale_kernel(const float* __restrict__ sW,
                                   float* __restrict__ Wn, int n, uint32_t seed) {
  int i = blockIdx.x * blockDim.x + threadIdx.x;
  if (i < n) Wn[i] = sW[i] * nrm(seed, (uint32_t)i);
}

// ---------------- LDS staging helper (async-to-LDS when available) ---------
__device__ __forceinline__ void stage_tile(const float* g, float* lds) {
#if __has_builtin(__builtin_amdgcn_global_load_async_to_lds_b128)
  __builtin_amdgcn_global_load_async_to_lds_b128(
      (__attribute__((address_space(1))) v4i_vs*)g,
      (__attribute__((address_space(3))) v4i_vs*)lds, 0, 0);
#else
  *(float4*)lds = *(const float4*)g;
#endif
}
__device__ __forceinline__ void stage_wait() {
#if __has_builtin(__builtin_amdgcn_global_load_async_to_lds_b128)
#if __has_builtin(__builtin_amdgcn_s_wait_asynccnt)
  __builtin_amdgcn_s_wait_asynccnt(0);
#else
  asm volatile("s_wait_asynccnt 0" ::: "memory");
#endif
#endif
}

// ---------------- WMMA dense / flipout layer -------------------------------
// Y[M,N] = act( X[M,K] @ W[N,K]^T + bias
//               + [FLIP] ((X*s_in) @ Wn^T + bn) * s_out ) (+ RES)
// Block = 256 thr = 8 waves; block tile = 128(M) x NB(N), NB = 16*NT.
// Wave tile = 16 x NB: NT f32 accumulators (+NT noise accumulators if FLIP).
// Weight slice W[NB x KC] (and Wn) staged in LDS, KC = 32.
template <int NT, bool FLIP>
__global__ __launch_bounds__(256)
void wmma_dense(const float* __restrict__ X, const float* __restrict__ W,
                const float* __restrict__ bias,
                const float* __restrict__ Wn, const float* __restrict__ bn,
                const float* __restrict__ RES, float* __restrict__ Y,
                int M, int K, int N, int act, int xrep_shift,
                uint32_t seed_in, uint32_t seed_out) {
  constexpr int KC = 32;
  constexpr int NB = 16 * NT;
  __shared__ float sW[64 * KC];
  __shared__ float sWn[64 * KC];

  const int nblks = N / NB;
  const int mblk  = blockIdx.x / nblks;
  const int nblk  = blockIdx.x - mblk * nblks;
  const int warp  = threadIdx.x >> 5;
  const int lane  = threadIdx.x & 31;
  const int half  = lane >> 4;
  const int l16   = lane & 15;

  const int tile_m_row = mblk * 128 + warp * 16;        // wave's 16 rows
  const bool mvalid    = (tile_m_row + 16) <= M;        // wave-uniform
  const int mrow_base  = mvalid ? tile_m_row : (M - 16);
  const int arow       = mrow_base + l16;               // logical A row
  const int xrow       = arow >> xrep_shift;            // physical X row
  const int ncol0      = nblk * NB;

  const float* xbase = X + (size_t)xrow * K + half * 2;

  v8f acc[NT];
  v8f acc2[NT];
#pragma unroll
  for (int t = 0; t < NT; ++t) { acc[t] = v8f{}; acc2[t] = v8f{}; }

  for (int k0 = 0; k0 < K; k0 += KC) {
    // ---- cooperative stage of W (and Wn) slice: NB x KC floats ----
    {
      const int elems = NB * KC;
      for (int e = threadIdx.x * 4; e < elems; e += 256 * 4) {
        const int r = e / KC, c = e - r * KC;
        const size_t goff = (size_t)(ncol0 + r) * K + k0 + c;
        stage_tile(W + goff, &sW[e]);
        if constexpr (FLIP) stage_tile(Wn + goff, &sWn[e]);
      }
      stage_wait();
    }
    __syncthreads();

    if (k0 + KC < K) __builtin_prefetch(xbase + k0 + KC, 0, 0);

    // ---- 8 k-steps of 4; NT (x2 if FLIP) WMMA per step ----
#pragma unroll
    for (int kk = 0; kk < KC; kk += 4) {
      v2f a;
      a.x = xbase[k0 + kk];
      a.y = xbase[k0 + kk + 1];
      v2f a2;
      if constexpr (FLIP) {
        const int kg = k0 + kk + half * 2;
        a2.x = a.x * rsign(seed_in, (uint32_t)(arow * K + kg));
        a2.y = a.y * rsign(seed_in, (uint32_t)(arow * K + kg + 1));
      }
#pragma unroll
      for (int t = 0; t < NT; ++t) {
        const float* bp = &sW[(t * 16 + l16) * KC + kk + half * 2];
        v2f b; b.x = bp[0]; b.y = bp[1];
        acc[t] = __builtin_amdgcn_wmma_f32_16x16x4_f32(false, a, false, b,
                                                       (short)0, acc[t],
                                                       false, false);
        if constexpr (FLIP) {
          const float* bp2 = &sWn[(t * 16 + l16) * KC + kk + half * 2];
          v2f b2; b2.x = bp2[0]; b2.y = bp2[1];
          acc2[t] = __builtin_amdgcn_wmma_f32_16x16x4_f32(false, a2, false, b2,
                                                          (short)0, acc2[t],
                                                          false, false);
        }
      }
    }
    __syncthreads();
  }

  if (!mvalid) return;  // wave-uniform; no barriers below

  // C/D layout: VGPR r -> (M = half*8 + r, N = l16)
#pragma unroll
  for (int t = 0; t < NT; ++t) {
    const int ncol = ncol0 + t * 16 + l16;
    const float bia = bias ? bias[ncol] : 0.0f;
    const float bnv = FLIP ? bn[ncol] : 0.0f;
#pragma unroll
    for (int r = 0; r < 8; ++r) {
      const int mrow = tile_m_row + half * 8 + r;
      float v = acc[t][r] + bia;
      if constexpr (FLIP)
        v += (acc2[t][r] + bnv) * rsign(seed_out, (uint32_t)(mrow * N + ncol));
      if (act == 1)      v = fmaxf(v, 0.0f);
      else if (act == 2) v = (v > 0.0f) ? v : 0.01f * v;
      if (RES) v += RES[(size_t)mrow * N + ncol];
      Y[(size_t)mrow * N + ncol] = v;
    }
  }
}

// ---------------- fused conv + relu + 2x2/2 maxpool ------------------------
__global__ void conv_pool_kernel(const float* __restrict__ in,
                                 const float* __restrict__ w,
                                 const float* __restrict__ bias,
                                 float* __restrict__ out,
                                 int Bn, int Cin, int Hin, int Win,
                                 int Cout, int KH, int KW, int stride, int pad,
                                 int Hc, int Wc, int Hp, int Wp, int ppad) {
  int idx = blockIdx.x * blockDim.x + threadIdx.x;
  int total = Bn * Cout * Hp * Wp;
  if (idx >= total) return;
  int wpx = idx % Wp; int t = idx / Wp;
  int hpx = t % Hp;   t /= Hp;
  int oc  = t % Cout; int b = t / Cout;

  float m = -INFINITY;
#pragma unroll
  for (int py = 0; py < 2; ++py) {
#pragma unroll
    for (int px = 0; px < 2; ++px) {
      int hc = hpx * 2 - ppad + py;
      int wc = wpx * 2 - ppad + px;
      if (hc < 0 || hc >= Hc || wc < 0 || wc >= Wc) continue;
      float s = bias[oc];
      int hi0 = hc * stride - pad, wi0 = wc * stride - pad;
      for (int ic = 0; ic < Cin; ++ic) {
        const float* ip  = in + ((size_t)(b * Cin + ic) * Hin) * Win;
        const float* wpt = w + ((size_t)(oc * Cin + ic) * KH) * KW;
        for (int kh = 0; kh < KH; ++kh) {
          int hi = hi0 + kh;
          if (hi < 0 || hi >= Hin) continue;
          for (int kw = 0; kw < KW; ++kw) {
            int wi = wi0 + kw;
            if (wi < 0 || wi >= Win) continue;
            s = fmaf(ip[hi * Win + wi], wpt[kh * KW + kw], s);
          }
        }
      }
      m = fmaxf(m, fmaxf(s, 0.0f));
    }
  }
  out[idx] = m;
}

// ---------------- mean over ensemble: (B,E,D) -> (B,D) ---------------------
__global__ void mean_e_kernel(const float* __restrict__ x,
                              float* __restrict__ o1, float* __restrict__ o2,
                              int D) {
  int b = blockIdx.x, d = threadIdx.x;
  const float* p = x + (size_t)b * ENS * D + d;
  float s = 0.0f;
#pragma unroll 8
  for (int e = 0; e < ENS; ++e) s += p[(size_t)e * D];
  s *= (1.0f / (float)ENS);
  o1[b * D + d] = s;
  if (o2) o2[b * D + d] = s;
}

// ---------------- observation-noise head: encoding -> R diag ---------------
__global__ void rdiag_kernel(const float* __restrict__ enc,
                             const float* __restrict__ W1, const float* __restrict__ b1,
                             const float* __restrict__ W2, const float* __restrict__ b2,
                             float* __restrict__ Rd) {
  __shared__ float h[32];
  int b = blockIdx.x, i = threadIdx.x;
  const float* e = enc + b * 64;
  float s = b1[i];
  for (int k = 0; k < 64; ++k) s = fmaf(e[k], W1[i * 64 + k], s);
  h[i] = fmaxf(s, 0.0f);
  __syncthreads();
  float d = b2[i];
  for (int k = 0; k < 32; ++k) d = fmaf(h[k], W2[i * 32 + k], d);
  const float cst  = 0.001f;
  const float init = 0.0948683298f;  // sqrt(0.1^2 - 0.001)
  d = d + cst;
  Rd[b * 32 + i] = d * d + init;
}

// ---------------- per-batch covariances ------------------------------------
__global__ void cov_kernel(const float* __restrict__ HX,
                           const float* __restrict__ Hm,
                           const float* __restrict__ SP,
                           const float* __restrict__ mA,
                           const float* __restrict__ Rd,
                           float* __restrict__ S, float* __restrict__ AH) {
  int b = blockIdx.x;
  int i = threadIdx.y, j = threadIdx.x;
  const float* hx = HX + (size_t)b * ENS * 32;
  const float* sp = SP + (size_t)b * ENS * 32;
  float hmi = Hm[b * 32 + i], hmj = Hm[b * 32 + j], mai = mA[b * 32 + i];
  float s = 0.0f, ah = 0.0f;
  for (int e = 0; e < ENS; ++e) {
    float haj = hx[e * 32 + j] - hmj;
    s  = fmaf(hx[e * 32 + i] - hmi, haj, s);
    ah = fmaf(sp[e * 32 + i] - mai, haj, ah);
  }
  s *= (1.0f / 127.0f);
  if (i == j) s += Rd[b * 32 + i];
  S[((size_t)b * 32 + i) * 32 + j]  = s;
  AH[((size_t)b * 32 + i) * 32 + j] = ah;  // /127 folded into K
}

// ---------------- per-batch 32x32 Gauss-Jordan inverse ---------------------
__global__ void inv32_kernel(const float* __restrict__ S, float* __restrict__ Si) {
  __shared__ float A[32][33];
  __shared__ float I_[32][33];
  int b = blockIdx.x;
  int i = threadIdx.y, j = threadIdx.x;
  A[i][j]  = S[((size_t)b * 32 + i) * 32 + j];
  I_[i][j] = (i == j) ? 1.0f : 0.0f;
  for (int p = 0; p < 32; ++p) {
    __syncthreads();
    float pivinv  = 1.0f / A[p][p];
    float rowp_j  = A[p][j];
    float irowp_j = I_[p][j];
    float f       = A[i][p];
    __syncthreads();
    if (i == p) {
      A[i][j]  = rowp_j * pivinv;
      I_[i][j] = irowp_j * pivinv;
    } else {
      float fp = f * pivinv;
      A[i][j]  = fmaf(-fp, rowp_j,  A[i][j]);
      I_[i][j] = fmaf(-fp, irowp_j, I_[i][j]);
    }
  }
  __syncthreads();
  Si[((size_t)b * 32 + i) * 32 + j] = I_[i][j];
}

// ---------------- K = AH @ Sinv / 127 --------------------------------------
__global__ void kmat_kernel(const float* __restrict__ AH,
                            const float* __restrict__ Si, float* __restrict__ Km) {
  int b = blockIdx.x;
  int i = threadIdx.y, j = threadIdx.x;
  const float* a = AH + (size_t)b * 1024;
  const float* s = Si + (size_t)b * 1024;
  float acc = 0.0f;
  for (int k = 0; k < 32; ++k) acc = fmaf(a[i * 32 + k], s[k * 32 + j], acc);
  Km[(size_t)b * 1024 + i * 32 + j] = acc * (1.0f / 127.0f);
}

// ---------------- state_new = state_pred + K @ (obs - H_X) -----------------
__global__ void gain_kernel(const float* __restrict__ SP, const float* __restrict__ Km,
                            const float* __restrict__ OBS, const float* __restrict__ HX,
                            float* __restrict__ OUT) {
  int idx = blockIdx.x * blockDim.x + threadIdx.x;  // B*E*32
  int i = idx & 31;
  int row = idx >> 5;
  int b = row >> 7;
  const float* kr = Km + (size_t)b * 1024 + i * 32;
  const float* ob = OBS + (size_t)row * 32;
  const float* hx = HX + (size_t)row * 32;
  float g = 0.0f;
#pragma unroll
  for (int j = 0; j < 32; ++j) g = fmaf(kr[j], ob[j] - hx[j], g);
  OUT[idx] = SP[idx] + g;
}

// ---------------- host-side dense dispatch ---------------------------------
static inline void launch_dense(hipStream_t s, const float* X, const float* W,
                                const float* bias, const float* Wn, const float* bn,
                                const float* RES, float* Y, int M, int K, int N,
                                int act, int xshift, uint32_t s_in, uint32_t s_out) {
  const int mblks = (M + 127) / 128;
  if (N % 64 == 0) {
    const int blocks = mblks * (N / 64);
    if (Wn)
      wmma_dense<4, true><<<blocks, 256, 0, s>>>(X, W, bias, Wn, bn, RES, Y,
                                                 M, K, N, act, xshift, s_in, s_out);
    else
      wmma_dense<4, false><<<blocks, 256, 0, s>>>(X, W, bias, nullptr, nullptr, RES, Y,
                                                  M, K, N, act, xshift, s_in, s_out);
  } else {  // N == 32
    const int blocks = mblks * (N / 32);
    if (Wn)
      wmma_dense<2, true><<<blocks, 256, 0, s>>>(X, W, bias, Wn, bn, RES, Y,
                                                 M, K, N, act, xshift, s_in, s_out);
    else
      wmma_dense<2, false><<<blocks, 256, 0, s>>>(X, W, bias, nullptr, nullptr, RES, Y,
                                                  M, K, N, act, xshift, s_in, s_out);
  }
}

// ===========================================================================
extern "C" void kernel_launch(void* const* d_in, const int* in_sizes, int n_in,
                              void* d_out, int out_size, void* d_ws, size_t ws_size,
                              hipStream_t stream) {
  // Input order: images, state_old, m_state, then params flattened jax-style
  // (dict keys alphabetical: om, on, pm, sm_b1, sm_b2, sm_conv, sm_lin1;
  //  flip dicts: muB, muW, sB, sW).
  const float* images    = (const float*)d_in[0];
  const float* state_old = (const float*)d_in[1];
  const float* omW[5] = {(const float*)d_in[3], (const float*)d_in[5],
                         (const float*)d_in[7], (const float*)d_in[9],
                         (const float*)d_in[11]};
  const float* omB[5] = {(const float*)d_in[4], (const float*)d_in[6],
                         (const float*)d_in[8], (const float*)d_in[10],
                         (const float*)d_in[12]};
  const float* onW1 = (const float*)d_in[13]; const float* onB1 = (const float*)d_in[14];
  const float* onW2 = (const float*)d_in[15]; const float* onB2 = (const float*)d_in[16];
  const float* pmMuB[4], *pmMuW[4], *pmSB[4], *pmSW[4];
  for (int l = 0; l < 4; ++l) {
    pmMuB[l] = (const float*)d_in[17 + 4 * l + 0];
    pmMuW[l] = (const float*)d_in[17 + 4 * l + 1];
    pmSB[l]  = (const float*)d_in[17 + 4 * l + 2];
    pmSW[l]  = (const float*)d_in[17 + 4 * l + 3];
  }
  const float* b1MuB = (const float*)d_in[33]; const float* b1MuW = (const float*)d_in[34];
  const float* b1SB  = (const float*)d_in[35]; const float* b1SW  = (const float*)d_in[36];
  const float* b2MuB = (const float*)d_in[37]; const float* b2MuW = (const float*)d_in[38];
  const float* b2SB  = (const float*)d_in[39]; const float* b2SW  = (const float*)d_in[40];
  const float* cW[3] = {(const float*)d_in[41], (const float*)d_in[43],
                        (const float*)d_in[45]};
  const float* cB[3] = {(const float*)d_in[42], (const float*)d_in[44],
                        (const float*)d_in[46]};
  const float* l1W = (const float*)d_in[47];
  const float* l1B = (const float*)d_in[48];

  float* out = (float*)d_out;
  float* o_state_new   = out + 0;
  float* o_m_state_new = out + 262144;
  float* o_m_state_prd = out + 264192;
  float* o_z           = out + 266240;
  float* o_obs         = out + 268288;
  float* o_hx_mean     = out + 530432;

  float* ws = (float*)d_ws;
  size_t off = 0;
  auto alloc = [&](size_t n) { float* p = ws + off; off += n; return p; };
  float* bufA  = alloc((size_t)8192 * 512);
  float* bufB  = alloc((size_t)8192 * 512);
  float* spred = alloc((size_t)8192 * 32);
  float* HX    = alloc((size_t)8192 * 32);
  float* enc   = alloc((size_t)8192 * 64);
  float* fbuf  = alloc((size_t)64 * 512);
  float* mA    = alloc(64 * 32);
  float* Hm    = alloc(64 * 32);
  float* encm  = alloc(64 * 64);
  float* Rd    = alloc(64 * 32);
  float* Smat  = alloc(64 * 1024);
  float* Simat = alloc(64 * 1024);
  float* AHmat = alloc(64 * 1024);
  float* Kmat  = alloc(64 * 1024);
  float* Wn[4]; float* bn[4];
  int pmIn[4]  = {32, 64, 512, 256};
  int pmOut[4] = {64, 512, 256, 32};
  for (int l = 0; l < 4; ++l) {
    Wn[l] = alloc((size_t)pmOut[l] * pmIn[l]);
    bn[l] = alloc(pmOut[l]);
  }
  float* WnB1 = alloc((size_t)64 * 512); float* bnB1 = alloc(64);
  float* WnB2 = alloc((size_t)32 * 64);  float* bnB2 = alloc(32);
  (void)ws_size; (void)in_sizes; (void)n_in; (void)out_size;

  auto nblk = [](int n, int b) { return (n + b - 1) / b; };

  // 1. noise-scaled weights
  for (int l = 0; l < 4; ++l) {
    int nw = pmOut[l] * pmIn[l];
    noise_scale_kernel<<<nblk(nw, 256), 256, 0, stream>>>(pmSW[l], Wn[l], nw, 0x1000u + l);
    noise_scale_kernel<<<1, 256, 0, stream>>>(pmSB[l], bn[l], pmOut[l], 0x2000u + l);
  }
  noise_scale_kernel<<<nblk(64 * 512, 256), 256, 0, stream>>>(b1SW, WnB1, 64 * 512, 0x3001u);
  noise_scale_kernel<<<1, 64, 0, stream>>>(b1SB, bnB1, 64, 0x3002u);
  noise_scale_kernel<<<nblk(32 * 64, 256), 256, 0, stream>>>(b2SW, WnB2, 32 * 64, 0x3003u);
  noise_scale_kernel<<<1, 32, 0, stream>>>(b2SB, bnB2, 32, 0x3004u);

  // 2. sensor conv stack
  conv_pool_kernel<<<nblk(64 * 16 * 55 * 55, 256), 256, 0, stream>>>(
      images, cW[0], cB[0], bufA, 64, 3, 224, 224, 16, 5, 5, 2, 1, 111, 111, 55, 55, 0);
  conv_pool_kernel<<<nblk(64 * 32 * 13 * 13, 256), 256, 0, stream>>>(
      bufA, cW[1], cB[1], bufB, 64, 16, 55, 55, 32, 5, 5, 2, 1, 27, 27, 13, 13, 0);
  conv_pool_kernel<<<nblk(64 * 64 * 7 * 7, 256), 256, 0, stream>>>(
      bufB, cW[2], cB[2], bufA, 64, 32, 13, 13, 64, 3, 3, 1, 1, 13, 13, 7, 7, 1);
  // lin1: (64,3136) -> (64,512), leaky
  launch_dense(stream, bufA, l1W, l1B, nullptr, nullptr, nullptr, fbuf,
               64, 3136, 512, 2, 0, 0, 0);

  // 3. process model (flipout chain)
  {
    const float* xin[4] = {state_old, bufA, bufB, bufA};
    float* xout[4]      = {bufA, bufB, bufA, spred};
    const float* res[4] = {nullptr, nullptr, nullptr, state_old};
    int act[4]          = {1, 1, 1, 0};
    for (int l = 0; l < 4; ++l)
      launch_dense(stream, xin[l], pmMuW[l], pmMuB[l], Wn[l], bn[l], res[l], xout[l],
                   8192, pmIn[l], pmOut[l], act[l], 0, 0x5100u + l, 0x5200u + l);
  }
  mean_e_kernel<<<64, 32, 0, stream>>>(spred, mA, o_m_state_prd, 32);

  // 4. observation model
  {
    int omIn[5]  = {32, 64, 128, 128, 64};
    int omOut[5] = {64, 128, 128, 64, 32};
    const float* xin[5] = {spred, bufA, bufB, bufA, bufB};
    float* xout[5]      = {bufA, bufB, bufA, bufB, HX};
    int act[5]          = {1, 1, 1, 1, 0};
    for (int l = 0; l < 5; ++l)
      launch_dense(stream, xin[l], omW[l], omB[l], nullptr, nullptr, nullptr, xout[l],
                   8192, omIn[l], omOut[l], act[l], 0, 0, 0);
  }
  mean_e_kernel<<<64, 32, 0, stream>>>(HX, Hm, o_hx_mean, 32);

  // 5. sensor flipout head
  launch_dense(stream, fbuf, b1MuW, b1MuB, WnB1, bnB1, nullptr, enc,
               8192, 512, 64, 2, /*xrep_shift=*/7, 0x6101u, 0x6201u);
  launch_dense(stream, enc, b2MuW, b2MuB, WnB2, bnB2, nullptr, o_obs,
               8192, 64, 32, 0, 0, 0x6102u, 0x6202u);
  mean_e_kernel<<<64, 32, 0, stream>>>(o_obs, o_z, nullptr, 32);
  mean_e_kernel<<<64, 64, 0, stream>>>(enc, encm, nullptr, 64);

  // 6. observation-noise head & EnKF algebra
  rdiag_kernel<<<64, 32, 0, stream>>>(encm, onW1, onB1, onW2, onB2, Rd);
  cov_kernel<<<64, dim3(32, 32), 0, stream>>>(HX, Hm, spred, mA, Rd, Smat, AHmat);
  inv32_kernel<<<64, dim3(32, 32), 0, stream>>>(Smat, Simat);
  kmat_kernel<<<64, dim3(32, 32), 0, stream>>>(AHmat, Simat, Kmat);

  // 7. gain + state update
  gain_kernel<<<nblk(8192 * 32, 256), 256, 0, stream>>>(spred, Kmat, o_obs, HX,
                                                        o_state_new);
  mean_e_kernel<<<64, 32, 0, stream>>>(o_state_new, o_m_state_new, nullptr, 32);
}